// Janossy_69836168233135
// MI455X (gfx1250) — compile-verified
//
#include <hip/hip_runtime.h>

typedef __attribute__((ext_vector_type(2))) float v2f;
typedef __attribute__((ext_vector_type(8))) float v8f;

#define DCOL 32  // feature dimension d (fixed by reference)

// Kernel 1: rank of each row by column 0 (stable: ties broken by index, matching
// jnp.argsort stability). Writes the two closed-form weight vectors:
//   wA[i] = n-1-rank(i)   (row i appears first in that many pairs)
//   wB[i] = rank(i)       (row i appears second in that many pairs)
__global__ void janossy_rank_kernel(const float* __restrict__ X,
                                    float* __restrict__ wA,
                                    float* __restrict__ wB, int n) {
    extern __shared__ float col0[];
    const int tid = threadIdx.x;
    const int nt  = blockDim.x;
    for (int j = tid; j < n; j += nt) col0[j] = X[j * DCOL];
    __syncthreads();
    const int gid = blockIdx.x * nt + tid;
    if (gid < n) {
        const float x = col0[gid];
        int rank = 0;
        for (int j = 0; j < n; ++j) {
            const float v = col0[j];
            rank += (int)((v < x) || (v == x && j < gid));
        }
        wA[gid] = (float)(n - 1 - rank);
        wB[gid] = (float)rank;
    }
}

// Kernel 2: out[2 x 32] = W[2 x n] * X[n x 32] via V_WMMA_F32_16X16X4_F32.
// A tile (16x4): row 0 = wA, row 1 = wB, rows 2..15 = 0 (mask-multiplied).
// One wave per 16-column N-tile; block of 64 threads = 2 waves covers d=32.
// A-layout (32-bit A 16x4): lanes 0..15 hold M=lane, VGPR0->K=k0, VGPR1->K=k0+1;
// lanes 16..31 hold M=lane-16, VGPR0->K=k0+2, VGPR1->K=k0+3.
// B/C/D: rows striped across lanes within a VGPR (C VGPR r: lanes0-15 -> M=r).
__global__ void janossy_wmma_kernel(const float* __restrict__ X,
                                    const float* __restrict__ wA,
                                    const float* __restrict__ wB,
                                    float* __restrict__ out, int n, float scale) {
    const int lane = threadIdx.x & 31;
    const int wave = threadIdx.x >> 5;   // N-tile id (0 or 1)
    const int half = lane >> 4;          // which K-pair this lane holds
    const int lm   = lane & 15;          // M index (A) / N index (B,C,D)
    const int ncol = wave * 16 + lm;

    // Branchless A sourcing: row-1 lanes read wB, everyone else reads wA
    // (always-valid address), then rows 2..15 are zeroed by mask multiply.
    const float* __restrict__ wsel = (lm == 1) ? wB : wA;
    const float amask = (lm < 2) ? 1.0f : 0.0f;

    v8f c = {};
#pragma unroll 4
    for (int k0 = 0; k0 < n; k0 += 4) {
        const int kx = k0 + 2 * half;
        v2f a;
        a.x = amask * wsel[kx];
        a.y = amask * wsel[kx + 1];
        v2f b;
        b.x = X[kx * DCOL + ncol];
        b.y = X[(kx + 1) * DCOL + ncol];
        // 8 args: (neg_a, A, neg_b, B, c_mod, C, reuse_a, reuse_b)
        c = __builtin_amdgcn_wmma_f32_16x16x4_f32(
                /*neg_a=*/false, a, /*neg_b=*/false, b,
                /*c_mod=*/(short)0, c, /*reuse_a=*/false, /*reuse_b=*/false);
    }

    // C VGPR0 lanes 0..15 -> M=0 (N=lane); C VGPR1 lanes 0..15 -> M=1.
    if (half == 0) {
        out[0 * DCOL + ncol] = c[0] * scale;
        out[1 * DCOL + ncol] = c[1] * scale;
    }
}

extern "C" void kernel_launch(void* const* d_in, const int* in_sizes, int n_in,
                              void* d_out, int out_size, void* d_ws, size_t ws_size,
                              hipStream_t stream) {
    (void)n_in; (void)out_size; (void)ws_size;
    const float* X = (const float*)d_in[0];
    // d_in[1] is k; reference fixes k=2 and the closed-form weights assume it.
    const int n = in_sizes[0] / DCOL;            // 2048

    float* wA  = (float*)d_ws;
    float* wB  = wA + n;
    float* out = (float*)d_out;

    const int threads = 256;
    const int blocks  = (n + threads - 1) / threads;
    janossy_rank_kernel<<<blocks, threads, n * sizeof(float), stream>>>(X, wA, wB, n);

    const float pairs = 0.5f * (float)n * (float)(n - 1); // C(n,2)
    janossy_wmma_kernel<<<1, 64, 0, stream>>>(X, wA, wB, out, n, 1.0f / pairs);
}